// SimplePredictionLayer_52785148068523
// MI455X (gfx1250) — compile-verified
//
#include <hip/hip_runtime.h>
#include <hip/hip_bf16.h>

typedef __attribute__((ext_vector_type(2))) float v2f;
typedef __attribute__((ext_vector_type(8))) float v8f;
typedef __attribute__((ext_vector_type(4))) int   v4i;

typedef __attribute__((address_space(1))) v4i g_v4i;   // global
typedef __attribute__((address_space(3))) v4i l_v4i;   // LDS

#define B_ 2
#define S_ 1024
#define N_ 128
#define D_ 512

#if defined(__AMDGCN__) && __has_builtin(__builtin_amdgcn_global_load_async_to_lds_b128)
#define HAVE_ASYNC_LDS 1
#else
#define HAVE_ASYNC_LDS 0
#endif

__device__ __forceinline__ void wait_asynccnt0() {
#if __has_builtin(__builtin_amdgcn_s_wait_asynccnt)
    __builtin_amdgcn_s_wait_asynccnt(0);
#else
    asm volatile("s_wait_asynccnt 0x0" ::: "memory");
#endif
}

// ---------------------------------------------------------------------------
// Kernel 0: out[i] = b2[0]  (GEMM kernel accumulates on top atomically)
// ---------------------------------------------------------------------------
__global__ void init_out_kernel(float* __restrict__ out, const float* __restrict__ b2) {
    int i = blockIdx.x * blockDim.x + threadIdx.x;
    if (i < B_ * N_) out[i] = b2[0];
}

// ---------------------------------------------------------------------------
// Kernel 1: sp[b,n,d] = max_s( mapping[b,s,n] * input[b,s,d] )
// One block per (b,n). Mask column staged in LDS; uniform branch skips
// zero-mask rows; sawZero reproduces the exact max-with-zero semantics.
// ---------------------------------------------------------------------------
__global__ void __launch_bounds__(256)
spmax_kernel(const float* __restrict__ mapping,
             const float* __restrict__ input,
             float* __restrict__ sp) {
    __shared__ float smask[S_];
    const int b = blockIdx.x / N_;
    const int n = blockIdx.x % N_;

    for (int s = threadIdx.x; s < S_; s += blockDim.x)
        smask[s] = mapping[((size_t)b * S_ + s) * N_ + n];
    __syncthreads();

    const int d = threadIdx.x * 2;                    // 256 threads cover D=512
    const float* inb = input + (size_t)b * S_ * D_ + d;

    float2 vmax = make_float2(-3.402823466e+38f, -3.402823466e+38f);
    bool sawZero = false;
    for (int s = 0; s < S_; ++s) {
        float m = smask[s];                           // scalar, uniform per block
        if (m != 0.0f) {
            float2 x = *(const float2*)(inb + (size_t)s * D_);
            vmax.x = fmaxf(vmax.x, m * x.x);
            vmax.y = fmaxf(vmax.y, m * x.y);
        } else {
            sawZero = true;
        }
    }
    if (sawZero) {                                    // zero-mask rows contribute 0
        vmax.x = fmaxf(vmax.x, 0.0f);
        vmax.y = fmaxf(vmax.y, 0.0f);
    }
    *(float2*)(sp + (size_t)blockIdx.x * D_ + d) = vmax;
}

// ---------------------------------------------------------------------------
// Kernel 2: out[r] += sum_n relu( (sp @ W1)[r,n] + b1[n] ) * W2[n]
// 64 blocks: blockIdx.x = mtile*4 + nchunk. Each block stages its 16x512 A
// strip in LDS via GLOBAL_LOAD_ASYNC_TO_LDS_B128 (ASYNCcnt-tracked, no VGPR
// round trip), 8 waves each own ONE 16-col N-tile. A re-staging is
// L2-resident (sp is 512KB << 192MB L2) so effectively free.
//
// Fragment layouts (wave32, ISA 7.12.2):
//   A 16x4 f32 : lane L -> m = L%16, VGPR v holds k = (L/16)*2 + v
//   B 4x16 f32 : lane L -> n = L%16, VGPR v holds k = (L/16)*2 + v
//   C/D 16x16  : lane L -> n = L%16, VGPR v holds m = v + (L/16)*8
// ---------------------------------------------------------------------------
__global__ void __launch_bounds__(256)
gemm_wmma_kernel(const float* __restrict__ sp,
                 const float* __restrict__ W1,
                 const float* __restrict__ b1,
                 const float* __restrict__ W2,
                 float* __restrict__ out) {
    __shared__ float As[16 * D_];                     // 32 KB
    const int mtile  = blockIdx.x >> 2;
    const int nchunk = blockIdx.x & 3;
    const int r0 = mtile * 16;

#if HAVE_ASYNC_LDS
    {
        // memory -> LDS direct, no VGPR round trip; tracked by ASYNCcnt
        char* g = (char*)(sp + (size_t)r0 * D_);      // const stripped for builtin
        char* l = (char*)As;
        for (int i = threadIdx.x; i < (16 * D_ * 4) / 16; i += blockDim.x) {
            __builtin_amdgcn_global_load_async_to_lds_b128(
                (g_v4i*)(g + (size_t)i * 16),
                (l_v4i*)(l + (size_t)i * 16),
                0, 0);
        }
        wait_asynccnt0();
    }
#else
    for (int i = threadIdx.x; i < 16 * D_; i += blockDim.x)
        As[i] = sp[(size_t)r0 * D_ + i];
#endif
    __syncthreads();

    const int wave = threadIdx.x >> 5;
    const int lane = threadIdx.x & 31;
    const int m    = lane & 15;
    const int half = lane >> 4;

    // one 16-column N-tile per wave (uniform, EXEC stays full)
    const int nt   = nchunk * 8 + wave;
    const int ncol = nt * 16 + m;

    v8f c = {};
    const float* arow = &As[m * D_ + half * 2];
    const float* bcol = &W1[(size_t)(half * 2) * D_ + ncol];

    for (int k = 0; k < D_; k += 4) {
        v2f a;
        a.x = arow[k];
        a.y = arow[k + 1];
        v2f bb;
        bb.x = bcol[(size_t)k * D_];
        bb.y = bcol[(size_t)(k + 1) * D_];
        c = __builtin_amdgcn_wmma_f32_16x16x4_f32(
                /*neg_a=*/false, a, /*neg_b=*/false, bb,
                /*c_mod=*/(short)0, c, /*reuse_a=*/false, /*reuse_b=*/false);
    }

    const float bias = b1[ncol];
    const float w2v  = W2[ncol];
    float part[8];
#pragma unroll
    for (int v = 0; v < 8; ++v) {
        float h = c[v] + bias;
        h = h > 0.0f ? h : 0.0f;                      // relu
        part[v] = h * w2v;                            // fuse @W2 matvec
    }

    // reduce over the 16 n-lanes inside each half-wave (masks <16 stay in-half)
#pragma unroll
    for (int off = 8; off >= 1; off >>= 1) {
#pragma unroll
        for (int v = 0; v < 8; ++v)
            part[v] += __shfl_xor(part[v], off, 32);
    }

    if (m == 0) {
#pragma unroll
        for (int v = 0; v < 8; ++v)
            atomicAdd(&out[r0 + half * 8 + v], part[v]);
    }
}

// ---------------------------------------------------------------------------
extern "C" void kernel_launch(void* const* d_in, const int* in_sizes, int n_in,
                              void* d_out, int out_size, void* d_ws, size_t ws_size,
                              hipStream_t stream) {
    const float* mapping = (const float*)d_in[0];   // [B,S,N]
    const float* input   = (const float*)d_in[1];   // [B,S,D]
    const float* W1      = (const float*)d_in[2];   // [D,D]
    const float* b1      = (const float*)d_in[3];   // [D]
    const float* W2      = (const float*)d_in[4];   // [D,1]
    const float* b2      = (const float*)d_in[5];   // [1]
    float* out = (float*)d_out;                     // [B,N,1] = 256 floats
    float* sp  = (float*)d_ws;                      // B*N*D floats = 512 KB scratch

    init_out_kernel<<<1, 256, 0, stream>>>(out, b2);
    spmax_kernel<<<B_ * N_, 256, 0, stream>>>(mapping, input, sp);
    gemm_wmma_kernel<<<(B_ * N_) / 16 * 4, 256, 0, stream>>>(sp, W1, b1, W2, out);
}